// MyModel_87522843560908
// MI455X (gfx1250) — compile-verified
//
#include <hip/hip_runtime.h>

typedef __attribute__((ext_vector_type(2))) float v2f;
typedef __attribute__((ext_vector_type(8))) float v8f;

#define QDIM 2048
#define VDIM 64
#define NNZV 65536
#define BHN  32

// ---------------------------------------------------------------------------
// Pass 1: zero the output (harness poisons d_out; atomics below accumulate).
// ---------------------------------------------------------------------------
__global__ void spmm_zero_out(float4* __restrict__ out, int n4) {
    int i = blockIdx.x * blockDim.x + threadIdx.x;
    if (i < n4) out[i] = float4{0.f, 0.f, 0.f, 0.f};
}

// ---------------------------------------------------------------------------
// Pass 2: COO scatter-SpMM via V_WMMA_F32_16X16X4_F32.
//
// One wave handles 16 nonzeros of one (b,h) slice.  Over t = 0..3 K-chunks we
// feed a diagonal-expanded A tile:  A_t[m,k] = val_{4t+k} iff m == 4t+k,
// and B_t[k,n] = b[col_{4t+k}, vbase+n].  Accumulating the 4 chunks yields
//   D[m,n] = val_m * b[col_m, vbase+n]   exactly (f32 WMMA, no rounding loss).
// V=64 -> 4 V-tiles of 16 columns, i.e. 16 WMMAs per 16 nonzeros.
//
// 32-bit operand layouts (ISA 7.12.2):
//   A 16x4 : lanes 0-15 hold A[m, 0..1] in vgpr.x/.y ; lanes 16-31 A[m, 2..3]
//   B 4x16 : lanes 0-15 hold B[0..1, n] in vgpr.x/.y ; lanes 16-31 B[2..3, n]
//   C/D    : vgpr g, lanes 0-15 -> D[g, n]; lanes 16-31 -> D[g+8, n]
// ---------------------------------------------------------------------------
__global__ __launch_bounds__(256) void spmm_wmma_scatter(
    const int*   __restrict__ aidx,   // [BH, NNZ, 2] (row, col)
    const float* __restrict__ avals,  // [BH, NNZ]
    const float* __restrict__ bmat,   // [BH, K=2048, V=64]
    float*       __restrict__ out)    // [BH, Q=2048, V=64]
{
    const int lane  = threadIdx.x & 31;
    const int wave  = threadIdx.x >> 5;
    const int w     = blockIdx.x * 8 + wave;       // global wave id
    const int bh    = w >> 12;                     // 4096 16-nnz tiles per bh
    const int tile  = w & 4095;
    const long base = (long)bh * NNZV + (long)tile * 16;

    // Lanes 0-15 own one nonzero each; others read via shuffles only.
    int row = 0, col = 0;
    float val = 0.f;
    if (lane < 16) {
        const int2 rc = ((const int2*)aidx)[base + lane];
        row = rc.x;
        col = rc.y;
        val = avals[base + lane];
    }

    const int r16 = lane & 15;       // N index (and M index for A)
    const int hi  = lane >> 4;       // which K half this lane carries

    const float* __restrict__ bb = bmat + (long)bh * (QDIM * (long)VDIM);

    v8f acc0 = {0,0,0,0,0,0,0,0};
    v8f acc1 = {0,0,0,0,0,0,0,0};
    v8f acc2 = {0,0,0,0,0,0,0,0};
    v8f acc3 = {0,0,0,0,0,0,0,0};

#pragma unroll
    for (int t = 0; t < 4; ++t) {
        const int j0 = 4 * t + 2 * hi;           // K = 2*hi     of this chunk
        const int j1 = j0 + 1;                   // K = 2*hi + 1

        const float vj0 = __shfl(val, j0, 32);
        const float vj1 = __shfl(val, j1, 32);
        const int   cj0 = __shfl(col, j0, 32);
        const int   cj1 = __shfl(col, j1, 32);

        v2f A;
        A.x = (r16 == j0) ? vj0 : 0.0f;          // diagonal expansion
        A.y = (r16 == j1) ? vj1 : 0.0f;

        const float* __restrict__ brow0 = bb + (long)cj0 * VDIM + r16;
        const float* __restrict__ brow1 = bb + (long)cj1 * VDIM + r16;

        v2f B0; B0.x = brow0[ 0]; B0.y = brow1[ 0];
        v2f B1; B1.x = brow0[16]; B1.y = brow1[16];
        v2f B2; B2.x = brow0[32]; B2.y = brow1[32];
        v2f B3; B3.x = brow0[48]; B3.y = brow1[48];

        acc0 = __builtin_amdgcn_wmma_f32_16x16x4_f32(false, A, false, B0,
                                                     (short)0, acc0, false, false);
        acc1 = __builtin_amdgcn_wmma_f32_16x16x4_f32(false, A, false, B1,
                                                     (short)0, acc1, false, false);
        acc2 = __builtin_amdgcn_wmma_f32_16x16x4_f32(false, A, false, B2,
                                                     (short)0, acc2, false, false);
        acc3 = __builtin_amdgcn_wmma_f32_16x16x4_f32(false, A, false, B3,
                                                     (short)0, acc3, false, false);
    }

    // Scatter: D[m, n] belongs to out[bh, row_m, vbase + n].
    float* __restrict__ ob = out + (long)bh * (QDIM * (long)VDIM);

#pragma unroll
    for (int g = 0; g < 8; ++g) {
        const int m  = g + 8 * hi;
        const int rm = __shfl(row, m, 32);
        float* dst = ob + (long)rm * VDIM + r16;
        atomicAdd(dst +  0, acc0[g]);
        atomicAdd(dst + 16, acc1[g]);
        atomicAdd(dst + 32, acc2[g]);
        atomicAdd(dst + 48, acc3[g]);
    }
}

// ---------------------------------------------------------------------------
extern "C" void kernel_launch(void* const* d_in, const int* in_sizes, int n_in,
                              void* d_out, int out_size, void* d_ws, size_t ws_size,
                              hipStream_t stream) {
    const int*   a_indices = (const int*)  d_in[0];   // [32, 65536, 2] i32
    const float* a_values  = (const float*)d_in[1];   // [32, 65536]    f32
    const float* b         = (const float*)d_in[2];   // [4,8,2048,64]  f32
    float*       out       = (float*)d_out;           // [4,8,2048,64]  f32

    // Zero pass: 32*2048*64 floats = 1,048,576 float4
    const int n4 = (BHN * QDIM * VDIM) / 4;
    spmm_zero_out<<<n4 / 256, 256, 0, stream>>>((float4*)out, n4);

    // SpMM pass: 32 bh * 4096 tiles = 131072 waves = 16384 blocks of 8 waves
    const int nblocks = (BHN * (NNZV / 16)) / 8;
    spmm_wmma_scatter<<<nblocks, 256, 0, stream>>>(a_indices, a_values, b, out);
}